// OnehotLSTM_15247133900807
// MI455X (gfx1250) — compile-verified
//
#include <hip/hip_runtime.h>
#include <hip/hip_bf16.h>

typedef __attribute__((ext_vector_type(2))) float v2f;
typedef __attribute__((ext_vector_type(8))) float v8f;

// ---------------------------------------------------------------------------
// Kernel 1: z[t][a*128+b][g] = sum_f x[b,f,a,t] * W_ih[g,f]  + bias(a,g)
// One wave per (b,a). 16 WMMA f32 16x16x4 per 16-wide t-tile, 32 t-tiles.
// x strides (floats): b:2097152, f:32768, a:512, t:1
// ---------------------------------------------------------------------------
__global__ __launch_bounds__(256)
void einsum_z_wmma(const float* __restrict__ x,
                   const float* __restrict__ W_ih,
                   const float* __restrict__ b_ih,
                   const float* __restrict__ b_hh,
                   float* __restrict__ z)
{
    const int lane  = threadIdx.x & 31;
    const int wave  = threadIdx.x >> 5;
    const int wid   = blockIdx.x * 8 + wave;     // 0..8191
    const int b     = wid >> 6;                  // 0..127
    const int a     = wid & 63;                  // 0..63

    const int n     = lane & 15;                 // tile column (t within tile / M row for A)
    const int khalf = (lane >> 4) << 1;          // 0 or 2 (K sub-pair)

    // ---- A-matrix fragments: W_ih rows g=0..3 in M rows 0..3; rows 4..15 zero
    const int m = n;
    v2f Afrag[16];
#pragma unroll
    for (int kc = 0; kc < 16; ++kc) {
        const int f0 = kc * 4 + khalf;
        float w0 = 0.0f, w1 = 0.0f;
        if (m < 4) {
            w0 = W_ih[m * 128 + f0];
            w1 = W_ih[m * 128 + f0 + 1];
        }
        Afrag[kc].x = w0;
        Afrag[kc].y = w1;
    }

    // ---- per-(a,g) bias, uniform across the wave
    const float bias0 = W_ih[0 * 128 + 64 + a] + b_ih[0] + b_hh[0];
    const float bias1 = W_ih[1 * 128 + 64 + a] + b_ih[1] + b_hh[1];
    const float bias2 = W_ih[2 * 128 + 64 + a] + b_ih[2] + b_hh[2];
    const float bias3 = W_ih[3 * 128 + 64 + a] + b_ih[3] + b_hh[3];

    const float* xb = x + (size_t)b * 2097152u + (size_t)a * 512u;
    const int ab = a * 128 + b;

    for (int tt = 0; tt < 32; ++tt) {
        const int t0 = tt * 16;
        v8f acc = {};
#pragma unroll
        for (int kc = 0; kc < 16; ++kc) {
            const int f0 = kc * 4 + khalf;
            const float* xp = xb + (size_t)f0 * 32768u + t0 + n;
            v2f Bfrag;
            Bfrag.x = xp[0];        // K = khalf,   N = t0+n  (t contiguous -> coalesced)
            Bfrag.y = xp[32768];    // K = khalf+1
            acc = __builtin_amdgcn_wmma_f32_16x16x4_f32(
                      false, Afrag[kc], false, Bfrag,
                      (short)0, acc, false, false);
        }
        // D: lane<16, VGPR r holds (g=r, t=t0+lane) -> one float4 per low lane
        if (lane < 16) {
            float4 o;
            o.x = acc[0] + bias0;
            o.y = acc[1] + bias1;
            o.z = acc[2] + bias2;
            o.w = acc[3] + bias3;
            *(float4*)(z + ((size_t)(t0 + lane) * 8192u + (size_t)ab) * 4u) = o;
        }
    }
}

// ---------------------------------------------------------------------------
// Kernel 2: independent scalar LSTM per (a,b); T=512 sequential steps.
// z stays in L2 (67 MB < 192 MB); coalesced float4 per step, next-step prefetch.
// ---------------------------------------------------------------------------
__device__ __forceinline__ float sigmoid_f(float v) {
    return 1.0f / (1.0f + __expf(-v));
}
__device__ __forceinline__ float tanh_f(float v) {
    return 1.0f - 2.0f / (1.0f + __expf(2.0f * v));
}

__global__ __launch_bounds__(256)
void lstm_scan(const float* __restrict__ z,
               const float* __restrict__ W_hh,
               float* __restrict__ h_out)
{
    const int idx = blockIdx.x * blockDim.x + threadIdx.x;   // ab = a*128+b
    const float w0 = W_hh[0], w1 = W_hh[1], w2 = W_hh[2], w3 = W_hh[3];

    float h = 0.0f, c = 0.0f;
    const float4* zp = (const float4*)z + idx;

    float4 znext = zp[0];
    for (int t = 0; t < 512; ++t) {
        const float4 zt = znext;
        if (t < 511) znext = zp[(size_t)(t + 1) * 8192u];    // prefetch next step
        const float ig = sigmoid_f(zt.x + h * w0);
        const float fg = sigmoid_f(zt.y + h * w1);
        const float gg = tanh_f   (zt.z + h * w2);
        const float og = sigmoid_f(zt.w + h * w3);
        c = fg * c + ig * gg;
        h = og * tanh_f(c);
    }
    h_out[idx] = h;
}

// ---------------------------------------------------------------------------
// Kernel 3: out[b, a] = softmax over a of h[a*128+b]. One block per b.
// ---------------------------------------------------------------------------
__global__ __launch_bounds__(64)
void softmax_rows(const float* __restrict__ h, float* __restrict__ out)
{
    __shared__ float sm[64];
    const int b = blockIdx.x;
    const int a = threadIdx.x;

    const float v = h[a * 128 + b];
    sm[a] = v;
    __syncthreads();
#pragma unroll
    for (int s = 32; s > 0; s >>= 1) {
        if (a < s) sm[a] = fmaxf(sm[a], sm[a + s]);
        __syncthreads();
    }
    const float mx = sm[0];
    __syncthreads();

    const float e = __expf(v - mx);
    sm[a] = e;
    __syncthreads();
#pragma unroll
    for (int s = 32; s > 0; s >>= 1) {
        if (a < s) sm[a] += sm[a + s];
        __syncthreads();
    }
    out[b * 64 + a] = e / sm[0];
}

// ---------------------------------------------------------------------------
extern "C" void kernel_launch(void* const* d_in, const int* in_sizes, int n_in,
                              void* d_out, int out_size, void* d_ws, size_t ws_size,
                              hipStream_t stream)
{
    const float* x    = (const float*)d_in[0];   // (128,64,64,512)
    const float* W_ih = (const float*)d_in[1];   // (4,128)
    const float* W_hh = (const float*)d_in[2];   // (4,1)
    const float* b_ih = (const float*)d_in[3];   // (4,)
    const float* b_hh = (const float*)d_in[4];   // (4,)
    float* out = (float*)d_out;                  // (128,64)

    float* z = (float*)d_ws;                           // 512*8192*4 floats = 64 MiB
    float* h = (float*)((char*)d_ws + (size_t)512 * 8192 * 4 * sizeof(float));

    // 8192 waves (one per (b,a)), 8 waves / 256-thread block
    einsum_z_wmma<<<1024, 256, 0, stream>>>(x, W_ih, b_ih, b_hh, z);
    // 8192 independent LSTM threads
    lstm_scan<<<32, 256, 0, stream>>>(z, W_hh, h);
    // softmax over A per batch row
    softmax_rows<<<128, 64, 0, stream>>>(h, out);
}